// FirstSelfAttention_4045859192938
// MI455X (gfx1250) — compile-verified
//
#include <hip/hip_runtime.h>
#include <hip/hip_bf16.h>

typedef __attribute__((ext_vector_type(2))) float v2f;
typedef __attribute__((ext_vector_type(8))) float v8f;
typedef __attribute__((ext_vector_type(4))) int   v4i;

// pointer-to-AS1 / pointer-to-AS3 v4i (attribute applies to pointee)
typedef __attribute__((address_space(1))) v4i* as1_v4i;
typedef __attribute__((address_space(3))) v4i* as3_v4i;

#define N_ATOMS 10000
#define N_PAIR  100000
#define F_DIM   128
#define N_HEADS 4
#define FH      32
#define N_DEG   3
#define M_TOT   15
#define K_RBF   32
#define DH_TOT  12      // N_DEG * N_HEADS
#define QK_DIM  384     // N_DEG * F

#if defined(__has_builtin)
#if __has_builtin(__builtin_amdgcn_global_load_async_to_lds_b128) && \
    __has_builtin(__builtin_amdgcn_s_wait_asynccnt)
#define USE_ASYNC_LDS 1
#endif
#endif
#ifndef USE_ASYNC_LDS
#define USE_ASYNC_LDS 0
#endif

static __device__ __forceinline__ v8f wmma4(v2f a, v2f b, v8f c) {
  // V_WMMA_F32_16X16X4_F32 : D = A(16x4,f32) * B(4x16,f32) + C(16x16,f32)
  return __builtin_amdgcn_wmma_f32_16x16x4_f32(
      false, a, false, b, (short)0, c, false, false);
}

// Stage a contiguous 16x128 f32 tile (8 KB) from global into LDS.
// Uses GLOBAL_LOAD_ASYNC_TO_LDS_B128 (ASYNCcnt) when available.
static __device__ __forceinline__ void stage_tile(const float* __restrict__ g,
                                                  float* lds_base, int tid) {
#if USE_ASYNC_LDS
  for (int c = tid; c < 512; c += 128) {   // 512 x 16B chunks
    __builtin_amdgcn_global_load_async_to_lds_b128(
        (as1_v4i)(g + c * 4),
        (as3_v4i)(lds_base + c * 4),
        0, 0);
  }
  __builtin_amdgcn_s_wait_asynccnt(0);
#else
  for (int i = tid; i < 16 * F_DIM; i += 128) lds_base[i] = g[i];
#endif
}

// ---------------- Kernel 1: x_f = x @ W_in + b_in ----------------
__global__ __launch_bounds__(128) void k_xf(const float* __restrict__ x,
                                            const float* __restrict__ W_in,
                                            const float* __restrict__ b_in,
                                            float* __restrict__ xf) {
  __shared__ float xs[16][F_DIM];
  const int tid = threadIdx.x;
  const int ab  = blockIdx.x * 16;
  stage_tile(x + (size_t)ab * F_DIM, &xs[0][0], tid);
  __syncthreads();
  const int lane  = tid & 31;
  const int wv    = tid >> 5;
  const int m     = lane & 15;
  const int ko    = (lane < 16) ? 0 : 2;
  const int rbase = (lane < 16) ? 0 : 8;
  for (int nt = wv; nt < 8; nt += 4) {
    const int col = nt * 16 + m;
    v8f c = {};
    for (int k0 = 0; k0 < F_DIM; k0 += 4) {
      v2f a, b;
      a.x = xs[m][k0 + ko];
      a.y = xs[m][k0 + ko + 1];
      b.x = W_in[(k0 + ko)     * F_DIM + col];
      b.y = W_in[(k0 + ko + 1) * F_DIM + col];
      c = wmma4(a, b, c);
    }
    const float bias = b_in[col];
#pragma unroll
    for (int r = 0; r < 8; ++r)
      xf[(ab + rbase + r) * F_DIM + col] = c[r] + bias;
  }
}

// ---------------- Kernel 2: Q/K = einsum('dhij,nhj->ndhi') ----------------
__global__ __launch_bounds__(128) void k_qk(const float* __restrict__ xf,
                                            const float* __restrict__ W_Q,
                                            const float* __restrict__ W_K,
                                            float* __restrict__ Q,
                                            float* __restrict__ K) {
  __shared__ float xs[16][F_DIM];
  const int tid = threadIdx.x;
  const int ab  = blockIdx.x * 16;
  stage_tile(xf + (size_t)ab * F_DIM, &xs[0][0], tid);
  __syncthreads();
  const int lane  = tid & 31;
  const int wv    = tid >> 5;
  const int m     = lane & 15;
  const int ko    = (lane < 16) ? 0 : 2;
  const int rbase = (lane < 16) ? 0 : 8;
  for (int task = wv; task < 48; task += 4) {
    const int qk  = task / 24;
    const int rem = task % 24;
    const int dh  = rem >> 1;       // d*4 + h
    const int nt  = rem & 1;
    const int h   = dh & 3;
    const float* W = qk ? W_K : W_Q;
    float*       O = qk ? K   : Q;
    const int icol = nt * 16 + m;   // output channel i
    v8f c = {};
    for (int k0 = 0; k0 < FH; k0 += 4) {
      v2f a, b;
      a.x = xs[m][h * FH + k0 + ko];
      a.y = xs[m][h * FH + k0 + ko + 1];
      // B[k][n=i] = W[d][h][i][k]  (contraction over j==k)
      b.x = W[(dh * FH + icol) * FH + k0 + ko];
      b.y = W[(dh * FH + icol) * FH + k0 + ko + 1];
      c = wmma4(a, b, c);
    }
#pragma unroll
    for (int r = 0; r < 8; ++r)
      O[(ab + rbase + r) * QK_DIM + dh * FH + icol] = c[r];
  }
}

// ---------------- Kernel 3: fused pair kernel ----------------
// h1 = ssp(rbf@W_f1+b), W_ij = (h1@W_f2+b)*phi, alpha = sum_f Qi*Kj*Wij,
// agg[i] += alpha_rep * sph (x) x_j     (atomics; agg fits in 192MB L2)
__global__ __launch_bounds__(256) void k_pair(
    const float* __restrict__ rbf,  const float* __restrict__ sph,
    const float* __restrict__ phi,  const int*   __restrict__ idx_i,
    const int*   __restrict__ idx_j,const float* __restrict__ mask,
    const float* __restrict__ W_f1, const float* __restrict__ b_f1,
    const float* __restrict__ W_f2, const float* __restrict__ b_f2,
    const float* __restrict__ Q,    const float* __restrict__ K,
    const float* __restrict__ xf,   float* __restrict__ agg) {
  __shared__ float h1s[16][K_RBF];
  __shared__ float wij[16][QK_DIM];
  __shared__ float alph[16][DH_TOT];
  __shared__ float sphs[16][M_TOT];
  __shared__ int   iis[16], jjs[16];
  __shared__ float msk[16];

  const int tid   = threadIdx.x;
  const int pb    = blockIdx.x * 16;
  const int lane  = tid & 31;
  const int wv    = tid >> 5;
  const int m     = lane & 15;
  const int ko    = (lane < 16) ? 0 : 2;
  const int rbase = (lane < 16) ? 0 : 8;

  if (tid < 16) {
    iis[tid] = idx_i[pb + tid];
    jjs[tid] = idx_j[pb + tid];
    msk[tid] = mask[pb + tid];
  }
  if (tid >= 16 && tid < 256) {
    int t = tid - 16;
    if (t < 16 * M_TOT)
      sphs[t / M_TOT][t % M_TOT] = sph[(pb + t / M_TOT) * M_TOT + (t % M_TOT)];
  }

  // phase 0: h1 = ssp(rbf@W_f1 + b_f1)    (wave 0 only; EXEC all-1s per wave)
  if (wv == 0) {
    for (int nt = 0; nt < 2; ++nt) {
      const int col = nt * 16 + m;
      v8f c = {};
      for (int k0 = 0; k0 < K_RBF; k0 += 4) {
        v2f a, b;
        a.x = rbf[(pb + m) * K_RBF + k0 + ko];
        a.y = rbf[(pb + m) * K_RBF + k0 + ko + 1];
        b.x = W_f1[(k0 + ko)     * K_RBF + col];
        b.y = W_f1[(k0 + ko + 1) * K_RBF + col];
        c = wmma4(a, b, c);
      }
      const float bias = b_f1[col];
#pragma unroll
      for (int r = 0; r < 8; ++r) {
        float v = c[r] + bias;
        // softplus(v) - ln2, numerically stable
        h1s[rbase + r][col] =
            fmaxf(v, 0.0f) + log1pf(__expf(-fabsf(v))) - 0.69314718055994531f;
      }
    }
  }
  __syncthreads();

  // phase 1: W_ij = (h1 @ W_f2 + b_f2) * phi_r_cut   (24 N-tiles over 8 waves)
  for (int t = wv * 3; t < wv * 3 + 3; ++t) {
    const int col = t * 16 + m;
    v8f c = {};
    for (int k0 = 0; k0 < K_RBF; k0 += 4) {
      v2f a, b;
      a.x = h1s[m][k0 + ko];
      a.y = h1s[m][k0 + ko + 1];
      b.x = W_f2[(k0 + ko)     * QK_DIM + col];
      b.y = W_f2[(k0 + ko + 1) * QK_DIM + col];
      c = wmma4(a, b, c);
    }
    const float bias = b_f2[col];
#pragma unroll
    for (int r = 0; r < 8; ++r) {
      const int row = rbase + r;
      wij[row][col] = (c[r] + bias) * phi[pb + row];
    }
  }
  __syncthreads();

  // phase 2: alpha[p,dh] = (1/sqrt(FH)) * m^2 * sum_f Qi*Kj*Wij
  if (tid < 16 * DH_TOT) {
    const int pp = tid & 15;
    const int dh = tid >> 4;
    const int i = iis[pp], j = jjs[pp];
    const float mm = msk[pp];
    float s = 0.f;
#pragma unroll 8
    for (int f = 0; f < FH; ++f)
      s += Q[(size_t)i * QK_DIM + dh * FH + f] *
           K[(size_t)j * QK_DIM + dh * FH + f] *
           wij[pp][dh * FH + f];
    alph[pp][dh] = s * mm * mm * 0.17677669529663687f; // 1/sqrt(32)
  }
  __syncthreads();

  // phase 3: scatter-add into agg (stays resident in L2)
  for (int w = tid; w < 16 * F_DIM; w += 256) {
    const int pp = w >> 7;
    const int f  = w & 127;
    const int h  = f >> 5;
    const int i = iis[pp], j = jjs[pp];
    const float xj = xf[(size_t)j * F_DIM + f] * msk[pp];
    float* dst = agg + (size_t)i * (M_TOT * F_DIM) + f;
#pragma unroll
    for (int mo = 0; mo < M_TOT; ++mo) {
      const int d = (mo < 3) ? 0 : ((mo < 8) ? 1 : 2);
      const float a = alph[pp][d * N_HEADS + h];
      atomicAdd(dst + mo * F_DIM, a * sphs[pp][mo] * xj);
    }
  }
}

// ---------------- Kernel 4: out = agg @ W_out + b_out ----------------
__global__ __launch_bounds__(128) void k_out(const float* __restrict__ agg,
                                             const float* __restrict__ W_out,
                                             const float* __restrict__ b_out,
                                             float* __restrict__ out) {
  __shared__ float as[16][F_DIM];
  const int tid = threadIdx.x;
  const size_t rb = (size_t)blockIdx.x * 16;
  stage_tile(agg + rb * F_DIM, &as[0][0], tid);
  __syncthreads();
  const int lane  = tid & 31;
  const int wv    = tid >> 5;
  const int m     = lane & 15;
  const int ko    = (lane < 16) ? 0 : 2;
  const int rbase = (lane < 16) ? 0 : 8;
  for (int nt = wv; nt < 8; nt += 4) {
    const int col = nt * 16 + m;
    v8f c = {};
    for (int k0 = 0; k0 < F_DIM; k0 += 4) {
      v2f a, b;
      a.x = as[m][k0 + ko];
      a.y = as[m][k0 + ko + 1];
      b.x = W_out[(k0 + ko)     * F_DIM + col];
      b.y = W_out[(k0 + ko + 1) * F_DIM + col];
      c = wmma4(a, b, c);
    }
    const float bias = b_out[col];
#pragma unroll
    for (int r = 0; r < 8; ++r)
      out[(rb + rbase + r) * F_DIM + col] = c[r] + bias;
  }
}

extern "C" void kernel_launch(void* const* d_in, const int* in_sizes, int n_in,
                              void* d_out, int out_size, void* d_ws, size_t ws_size,
                              hipStream_t stream) {
  const float* x       = (const float*)d_in[0];
  const float* rbf_ij  = (const float*)d_in[1];
  const float* sph_ij  = (const float*)d_in[2];
  const float* phi     = (const float*)d_in[3];
  const int*   idx_i   = (const int*)  d_in[4];
  const int*   idx_j   = (const int*)  d_in[5];
  const float* mask    = (const float*)d_in[6];
  const float* W_Q     = (const float*)d_in[7];
  const float* W_K     = (const float*)d_in[8];
  const float* W_in    = (const float*)d_in[9];
  const float* b_in    = (const float*)d_in[10];
  const float* W_f1    = (const float*)d_in[11];
  const float* b_f1    = (const float*)d_in[12];
  const float* W_f2    = (const float*)d_in[13];
  const float* b_f2    = (const float*)d_in[14];
  const float* W_out   = (const float*)d_in[15];
  const float* b_out   = (const float*)d_in[16];
  float* out = (float*)d_out;

  // workspace layout
  float* xf  = (float*)d_ws;                               // 10000*128
  float* Q   = xf  + (size_t)N_ATOMS * F_DIM;              // 10000*384
  float* K   = Q   + (size_t)N_ATOMS * QK_DIM;             // 10000*384
  float* agg = K   + (size_t)N_ATOMS * QK_DIM;             // 10000*15*128

  (void)hipMemsetAsync(agg, 0, (size_t)N_ATOMS * M_TOT * F_DIM * sizeof(float),
                       stream);

  k_xf  <<<N_ATOMS / 16, 128, 0, stream>>>(x, W_in, b_in, xf);
  k_qk  <<<N_ATOMS / 16, 128, 0, stream>>>(xf, W_Q, W_K, Q, K);
  k_pair<<<N_PAIR  / 16, 256, 0, stream>>>(rbf_ij, sph_ij, phi, idx_i, idx_j,
                                           mask, W_f1, b_f1, W_f2, b_f2,
                                           Q, K, xf, agg);
  k_out <<<(N_ATOMS * M_TOT) / 16, 128, 0, stream>>>(agg, W_out, b_out, out);
}